// RPSRoIAlign_82016695485110
// MI455X (gfx1250) — compile-verified
//
#include <hip/hip_runtime.h>
#include <math.h>

// RPSRoIAlign for MI455X (gfx1250, wave32).
// One wave per (roi, bin). Lanes cooperatively gather bilinear corner values
// into the A-matrix layout of V_WMMA_F32_16X16X4_F32 (M = output channel,
// K = sample*4 + corner, chained over 4 K-chunks); B holds the shared bilinear
// weights (channel-independent -> naturally broadcast across N columns).
// D column 0 carries the 10 reduced channel outputs in full f32 precision.

typedef float v2f __attribute__((ext_vector_type(2)));
typedef float v8f __attribute__((ext_vector_type(8)));

constexpr int   kGroup  = 7;
constexpr int   kPooled = 7;
constexpr int   kBins   = kPooled * kPooled;     // 49
constexpr int   kOC     = 10;
constexpr int   kC      = kOC * kGroup * kGroup; // 490
constexpr int   kH      = 256;
constexpr int   kW      = 256;
constexpr float kScale  = 0.25f;

__global__ __launch_bounds__(256)
void rpsroi_align_wmma(const float* __restrict__ feat,
                       const float* __restrict__ rois,
                       float* __restrict__ out,
                       int n_rois)
{
    const int wave = blockIdx.x * (blockDim.x >> 5) + (threadIdx.x >> 5);
    if (wave >= n_rois * kBins) return;              // wave-uniform: EXEC stays all-1s
    const int lane = threadIdx.x & 31;

    const int roi_id = wave / kBins;
    const int bin    = wave - roi_id * kBins;
    const int ph     = bin / kPooled;
    const int pw     = bin - ph * kPooled;

    // RoI parameters (wave-uniform; compiler scalarizes these loads).
    const float* r  = rois + (size_t)roi_id * 6;
    const int    b  = (int)r[0];
    const float  cx = r[1] * kScale;
    const float  cy = r[2] * kScale;
    const float  w  = fmaxf(r[3] * kScale, 0.1f);
    const float  h  = fmaxf(r[4] * kScale, 0.1f);
    const float  th = r[5];
    const float  binw = w / (float)kPooled;
    const float  binh = h / (float)kPooled;
    float sth, cth;
    sincosf(th, &sth, &cth);

    // Lane -> (M row, corner pair).  A 16x4 f32 layout: lanes 0-15 hold M=lane,
    // VGPR0 = K even, VGPR1 = K odd; lanes 16-31 hold the upper K pair.
    const int m    = lane & 15;
    const int half = lane >> 4;                      // 0: corners (y0,x0/x1), 1: (y1,x0/x1)
    const int ch   = (m < kOC) ? m : (kOC - 1);      // clamp dead rows onto hot lines
    const int cin  = (ch * kGroup + ph) * kGroup + pw;
    const float* fb = feat + ((size_t)(b * kC + cin)) * (size_t)(kH * kW);

    v8f acc = {};
    #pragma unroll
    for (int t = 0; t < 4; ++t) {                    // t = sample index (sy*2 + sx)
        const int   sy = t >> 1, sx = t & 1;
        const float oy = ((float)ph + ((float)sy + 0.5f) * 0.5f) * binh - h * 0.5f;
        const float ox = ((float)pw + ((float)sx + 0.5f) * 0.5f) * binw - w * 0.5f;
        const float X  = cx + ox * cth - oy * sth;
        const float Y  = cy + ox * sth + oy * cth;
        const bool valid = (Y >= -1.0f) && (Y <= (float)kH) &&
                           (X >= -1.0f) && (X <= (float)kW);
        const float Xc = fminf(fmaxf(X, 0.0f), (float)(kW - 1));
        const float Yc = fminf(fmaxf(Y, 0.0f), (float)(kH - 1));
        const int   x0 = (int)floorf(Xc);
        const int   y0 = (int)floorf(Yc);
        const int   x1 = min(x0 + 1, kW - 1);
        const int   y1 = min(y0 + 1, kH - 1);
        const float lx = Xc - (float)x0, ly = Yc - (float)y0;
        const float hx = 1.0f - lx,      hy = 1.0f - ly;

        const int   yy = half ? y1 : y0;             // this lane's corner row
        const float wy = half ? ly : hy;
        const float s  = valid ? 0.25f : 0.0f;       // fold sample-mean + validity

        v2f a, bw;                                   // adjacent addrs: same cacheline
        a.x  = fb[yy * kW + x0];
        a.y  = fb[yy * kW + x1];
        bw.x = wy * hx * s;                          // weight for corner (yy, x0)
        bw.y = wy * lx * s;                          // weight for corner (yy, x1)

        // D += A(16x4 f32) x B(4x16 f32); weights identical across lanes of a
        // half -> every D column equals the desired per-channel reduction.
        acc = __builtin_amdgcn_wmma_f32_16x16x4_f32(
                  /*neg_a=*/false, a, /*neg_b=*/false, bw,
                  /*c_mod=*/(short)0, acc, /*reuse_a=*/false, /*reuse_b=*/false);
    }

    // D column 0: lane 0 -> M=0..7 (VGPR 0..7), lane 16 -> M=8..15 (VGPR 0..7).
    float* o = out + (size_t)roi_id * kOC * kBins + bin;
    if (lane == 0) {
        #pragma unroll
        for (int q = 0; q < 8; ++q)
            o[(size_t)q * kBins] = acc[q];
    } else if (lane == 16) {
        #pragma unroll
        for (int q = 0; q < kOC - 8; ++q)
            o[(size_t)(8 + q) * kBins] = acc[q];
    }
}

extern "C" void kernel_launch(void* const* d_in, const int* in_sizes, int n_in,
                              void* d_out, int out_size, void* d_ws, size_t ws_size,
                              hipStream_t stream) {
    const float* features = (const float*)d_in[0];
    const float* rois     = (const float*)d_in[1];
    float*       out      = (float*)d_out;

    const int n_rois      = in_sizes[1] / 6;
    const int total_waves = n_rois * kBins;          // one wave per (roi, bin)
    const int waves_per_block = 256 / 32;
    const int grid = (total_waves + waves_per_block - 1) / waves_per_block;

    rpsroi_align_wmma<<<grid, 256, 0, stream>>>(features, rois, out, n_rois);
}